// DataEmbedding_20985210208419
// MI455X (gfx1250) — compile-verified
//
#include <hip/hip_runtime.h>
#include <hip/hip_bf16.h>

typedef __attribute__((ext_vector_type(2))) float v2f;
typedef __attribute__((ext_vector_type(8))) float v8f;

#define C_IN      7
#define D_MODEL   512
#define KTAPS     8
#define NUM_KER   74
#define PROJ_IN   512      // 73*7 + 1
#define L_SEQ     4096
#define BM        128      // rows per block
#define BN        256      // cols per block
#define BK        32       // K chunk
#define A_PITCH   36       // even pitch: 8B-aligned frags, banks 4*(9r%16)+kk distinct
#define B_PITCH   257      // odd pitch: fill bank = (j+n)%64, conflict-free
#define SIN_COEF  (-0.0359778925f)   // -2*ln(10000)/512

__global__ __launch_bounds__(256)
void data_embedding_fused(const float* __restrict__ x,
                          const int*   __restrict__ x_mark,
                          const float* __restrict__ kern,
                          const float* __restrict__ proj_w,
                          const float* __restrict__ proj_b,
                          float*       __restrict__ out)
{
    __shared__ __align__(16) float As[BM * A_PITCH];   // combined chunk (computed)
    __shared__ __align__(16) float Bs[BK * B_PITCH];   // proj_w^T chunk
    __shared__ float xl[135 * C_IN];            // x rows l0-7 .. l0+127 (zero-padded)
    __shared__ float kr[NUM_KER * KTAPS];       // conv kernels
    __shared__ float tsm[7 * D_MODEL];          // sinusoid rows 0..6 (temporal)
    __shared__ float pb[D_MODEL];               // proj bias
    __shared__ int   xm[BM * 4];                // x_mark for this row block

    const int tid  = threadIdx.x;
    const int bid  = blockIdx.x;
    const int mblk = bid >> 1;                  // 0..1023
    const int n0   = (bid & 1) * BN;            // 0 or 256
    const int b    = mblk >> 5;                 // 32 row-blocks per batch elem
    const int l0   = (mblk & 31) * BM;

    // ---------------- prologue: stage invariants ----------------
    for (int i = tid; i < NUM_KER * KTAPS; i += 256) kr[i] = kern[i];
    for (int i = tid; i < D_MODEL; i += 256)         pb[i] = proj_b[i];
    for (int i = tid; i < BM * 4; i += 256)
        xm[i] = x_mark[(size_t)(b * L_SEQ + l0) * 4 + i];
    for (int i = tid; i < 135 * C_IN; i += 256) {
        int rr = i / C_IN, c = i - rr * C_IN;
        int l  = l0 - 7 + rr;
        xl[i]  = (l >= 0) ? x[((size_t)b * L_SEQ + l) * C_IN + c] : 0.0f;
    }
    for (int i = tid; i < 7 * D_MODEL; i += 256) {
        int p = i >> 9;
        int d = i & (D_MODEL - 1);
        float div = __expf((float)(d >> 1) * SIN_COEF);
        float ang = (float)p * div;
        tsm[i] = (d & 1) ? __cosf(ang) : __sinf(ang);
    }

    const int lane = tid & 31;
    const int wid  = tid >> 5;
    const int wm   = wid >> 1;       // 0..3 -> 32-row strip
    const int wn   = wid & 1;        // 0..1 -> 128-col strip
    const int lrow = lane & 15;
    const int half = lane >> 4;      // K half-select for f32 WMMA frags

    v8f acc[2][8];
    const v8f vzero = {0.f, 0.f, 0.f, 0.f, 0.f, 0.f, 0.f, 0.f};
    #pragma unroll
    for (int mt = 0; mt < 2; ++mt)
        #pragma unroll
        for (int nt = 0; nt < 8; ++nt)
            acc[mt][nt] = vzero;

    // ---------------- K loop: fused conv -> WMMA GEMM ----------------
    for (int kc = 0; kc < PROJ_IN; kc += BK) {
        __syncthreads();
        // A chunk: combined[l0+r, kc+jl] = sum_m x[l0+r+m-7, c] * ker[k, m]
        #pragma unroll
        for (int it = 0; it < (BM * BK) / 256; ++it) {
            int i  = tid + it * 256;
            int r  = i >> 5;
            int jl = i & (BK - 1);
            int j  = kc + jl;
            int k, c;
            if (j < 511) { k = j / 7; c = j - k * 7; } else { k = 73; c = 0; }
            const float* xr = &xl[r * C_IN + c];
            const float* kw = &kr[k * KTAPS];
            float s = 0.f;
            #pragma unroll
            for (int m = 0; m < KTAPS; ++m)
                s = __builtin_fmaf(xr[m * C_IN], kw[m], s);
            As[r * A_PITCH + jl] = s;
        }
        // B chunk: Bs[jl][n] = proj_w[n0+n][kc+jl], float4 along contiguous j
        #pragma unroll
        for (int it = 0; it < (BK * BN) / (256 * 4); ++it) {
            int i   = tid + it * 256;
            int jl4 = (i & 7) * 4;
            int n   = i >> 3;
            float4 w = *reinterpret_cast<const float4*>(
                &proj_w[(size_t)(n0 + n) * PROJ_IN + (kc + jl4)]);
            Bs[(jl4 + 0) * B_PITCH + n] = w.x;
            Bs[(jl4 + 1) * B_PITCH + n] = w.y;
            Bs[(jl4 + 2) * B_PITCH + n] = w.z;
            Bs[(jl4 + 3) * B_PITCH + n] = w.w;
        }
        if (kc + BK < PROJ_IN)
            __builtin_prefetch(&proj_w[(size_t)(n0 + tid) * PROJ_IN + kc + BK], 0, 1);
        __syncthreads();

        #pragma unroll
        for (int ks = 0; ks < BK / 4; ++ks) {
            const int kk = ks * 4 + half * 2;   // lanes 0-15: K=kk,kk+1; 16-31: +2
            v2f a[2], bf[8];
            #pragma unroll
            for (int mt = 0; mt < 2; ++mt) {
                int row = wm * 32 + mt * 16 + lrow;
                a[mt] = *reinterpret_cast<const v2f*>(&As[row * A_PITCH + kk]);
            }
            #pragma unroll
            for (int nt = 0; nt < 8; ++nt) {
                int n = wn * 128 + nt * 16 + lrow;
                bf[nt].x = Bs[kk * B_PITCH + n];
                bf[nt].y = Bs[(kk + 1) * B_PITCH + n];
            }
            #pragma unroll
            for (int mt = 0; mt < 2; ++mt)
                #pragma unroll
                for (int nt = 0; nt < 8; ++nt)
                    acc[mt][nt] = __builtin_amdgcn_wmma_f32_16x16x4_f32(
                        false, a[mt], false, bf[nt], (short)0, acc[mt][nt],
                        false, false);
        }
    }

    // ---------------- epilogue: bias + positional + temporal ----------------
    float divv[8], pbv[8];
    int   dv[8];
    #pragma unroll
    for (int nt = 0; nt < 8; ++nt) {
        int d    = n0 + wn * 128 + nt * 16 + lrow;
        dv[nt]   = d;
        divv[nt] = __expf((float)(d >> 1) * SIN_COEF);
        pbv[nt]  = pb[d];
    }
    const bool odd = (lrow & 1) != 0;   // column parity is lane-invariant per tile

    #pragma unroll
    for (int mt = 0; mt < 2; ++mt) {
        #pragma unroll
        for (int v = 0; v < 8; ++v) {
            int rblk = wm * 32 + mt * 16 + half * 8 + v;  // C layout: VGPR v -> row v / v+8
            int l    = l0 + rblk;
            size_t grow = (size_t)b * L_SEQ + l;
            int i0 = xm[rblk * 4 + 0];
            int i1 = xm[rblk * 4 + 1];
            int i2 = xm[rblk * 4 + 2];
            int i3 = xm[rblk * 4 + 3];
            #pragma unroll
            for (int nt = 0; nt < 8; ++nt) {
                int d = dv[nt];
                float sn, cs;
                __sincosf((float)l * divv[nt], &sn, &cs);
                float pos = odd ? cs : sn;
                float val = acc[mt][nt][v] + pbv[nt] + pos
                          + tsm[i0 * D_MODEL + d] + tsm[i1 * D_MODEL + d]
                          + tsm[i2 * D_MODEL + d] + tsm[i3 * D_MODEL + d];
                out[grow * D_MODEL + d] = val;
            }
        }
    }
}

extern "C" void kernel_launch(void* const* d_in, const int* in_sizes, int n_in,
                              void* d_out, int out_size, void* d_ws, size_t ws_size,
                              hipStream_t stream) {
    const float* x      = (const float*)d_in[0];
    const int*   x_mark = (const int*)  d_in[1];
    const float* kern   = (const float*)d_in[2];
    const float* proj_w = (const float*)d_in[3];
    const float* proj_b = (const float*)d_in[4];
    float*       out    = (float*)d_out;

    // grid: (B*L/BM) row-blocks * (D_MODEL/BN) col-halves = 1024 * 2
    dim3 grid(2048), block(256);
    hipLaunchKernelGGL(data_embedding_fused, grid, block, 0, stream,
                       x, x_mark, kern, proj_w, proj_b, out);
}